// Logistic_loss_57337813401632
// MI455X (gfx1250) — compile-verified
//
#include <hip/hip_runtime.h>
#include <hip/hip_bf16.h>
#include <math.h>

typedef __attribute__((ext_vector_type(2))) float v2f;
typedef __attribute__((ext_vector_type(8))) float v8f;

#define N_ROWS 100000
#define N_CLS  20
#define DIM    2001
#define REG_C  0.001f

#define KSTEPS 501                 // ceil(2004/4): covers k in [0,2004), zero-padded past 2000
#define TILES  (N_ROWS / 16)       // 6250 row-tiles of 16
#define WPB    8                   // waves per block (wave32)
#define BLKT   256
#define MAIN_BLOCKS ((TILES + WPB - 1) / WPB)   // 782

// Workspace layout (floats):
//   [0, PACK_FLOATS)                : packed B operands (2 class-tiles x 501 ksteps x 32 lanes x 2)
//   [PACK_FLOATS, +MAIN_BLOCKS)     : per-block partial loss sums
#define PACK_FLOATS (2 * KSTEPS * 32 * 2)       // 64128 floats ~= 257 KB

// ---------------------------------------------------------------------------
// Pack weights [20][2001] into WMMA B-operand lane layout, zero-padded to
// 32 classes x 2004 k.  Layout: Bp[((tile*KSTEPS + ks)*32 + lane)*2 + j]
// holds W[tile*16 + (lane&15)][4*ks + 2*(lane>>4) + j]  (or 0 if OOB).
// ---------------------------------------------------------------------------
__global__ void pack_weights_kernel(const float* __restrict__ W,
                                    float* __restrict__ Bp) {
    int t = blockIdx.x * blockDim.x + threadIdx.x;
    if (t >= PACK_FLOATS) return;
    int j     = t & 1;
    int lane  = (t >> 1) & 31;
    int rest  = t >> 6;                  // tile*KSTEPS + kstep
    int kstep = rest % KSTEPS;
    int tile  = rest / KSTEPS;
    int c = tile * 16 + (lane & 15);
    int k = 4 * kstep + 2 * (lane >> 4) + j;
    float v = 0.0f;
    if (c < N_CLS && k < DIM) v = W[c * DIM + k];
    Bp[t] = v;
}

// ---------------------------------------------------------------------------
// Main kernel: each wave owns one 16-row tile; K-loop of V_WMMA_F32_16X16X4_F32
// (two class-tiles), then per-row loss via LDS, block partial sum in fixed order.
// ---------------------------------------------------------------------------
__global__ void __launch_bounds__(BLKT)
logistic_loss_main_kernel(const float* __restrict__ X,
                          const float* __restrict__ Bp,
                          const int*   __restrict__ Y,
                          float*       __restrict__ partials) {
    __shared__ float sc[WPB][16][21];        // scores [row][class], padded stride
    __shared__ float lossBuf[WPB * 16];

    const int lane = threadIdx.x & 31;
    const int w    = threadIdx.x >> 5;
    const int tile = blockIdx.x * WPB + w;
    const bool valid = (tile < TILES);       // wave-uniform

    if (lane < 16) lossBuf[w * 16 + lane] = 0.0f;

    if (valid) {
        const int half = lane >> 4;          // selects K sub-pair per A/B layout
        const int idx  = lane & 15;          // row (A) / class (B,C,D) within tile
        const size_t rowBase = (size_t)(tile * 16 + idx) * DIM;
        const v2f* __restrict__ bv = (const v2f*)Bp;   // 8B-aligned (hipMalloc ws)

        v8f c0 = {};   // classes 0..15
        v8f c1 = {};   // classes 16..31 (20..31 are zero columns)

        #pragma unroll 4
        for (int ks = 0; ks < 500; ++ks) {
            const int ka = 4 * ks + 2 * half;          // <= 1999, always in-bounds
            v2f a;
            a.x = X[rowBase + ka];
            a.y = X[rowBase + ka + 1];
            v2f b0 = bv[(size_t)ks * 32 + lane];
            v2f b1 = bv[(size_t)(KSTEPS + ks) * 32 + lane];
            c0 = __builtin_amdgcn_wmma_f32_16x16x4_f32(false, a, false, b0,
                                                       (short)0, c0, false, false);
            c1 = __builtin_amdgcn_wmma_f32_16x16x4_f32(false, a, false, b1,
                                                       (short)0, c1, false, false);
        }
        {   // epilogue ks = 500: only k == 2000 is a real X element
            v2f a;
            a.x = (half == 0) ? X[rowBase + 2000] : 0.0f;
            a.y = 0.0f;
            v2f b0 = bv[(size_t)500 * 32 + lane];
            v2f b1 = bv[(size_t)(KSTEPS + 500) * 32 + lane];
            c0 = __builtin_amdgcn_wmma_f32_16x16x4_f32(false, a, false, b0,
                                                       (short)0, c0, false, false);
            c1 = __builtin_amdgcn_wmma_f32_16x16x4_f32(false, a, false, b1,
                                                       (short)0, c1, false, false);
        }

        // Scatter D fragment to LDS: lane = class, VGPR j = row (+8 for upper half)
        #pragma unroll
        for (int j = 0; j < 8; ++j) {
            const int row = j + 8 * half;
            sc[w][row][idx] = c0[j];
            if (idx < 4) sc[w][row][16 + idx] = c1[j];
        }
        // Same-wave LDS RAW is in-order; no cross-wave sharing -> no barrier needed.

        if (lane < 16) {
            const int y = Y[tile * 16 + lane];
            float m = -1e30f;
            #pragma unroll
            for (int c = 0; c < N_CLS; ++c) m = fmaxf(m, -sc[w][lane][c]);
            float se = 0.0f;
            #pragma unroll
            for (int c = 0; c < N_CLS; ++c) se += __expf(-sc[w][lane][c] - m);
            // loss_i = scores[i, y] + logsumexp(-scores[i, :])
            lossBuf[w * 16 + lane] = sc[w][lane][y] + m + __logf(se);
        }
    }

    __syncthreads();
    if (threadIdx.x == 0) {
        float s = 0.0f;
        for (int i = 0; i < WPB * 16; ++i) s += lossBuf[i];   // fixed order
        partials[blockIdx.x] = s;
    }
}

// ---------------------------------------------------------------------------
// Finalize: mean of partials + REG * frobenius_norm(init_weights). Deterministic
// fixed-order reductions only.
// ---------------------------------------------------------------------------
__global__ void __launch_bounds__(BLKT)
finalize_kernel(const float* __restrict__ partials, int nPart,
                const float* __restrict__ initW, int iwCount,
                float* __restrict__ out) {
    __shared__ float red[BLKT];
    float s = 0.0f;
    for (int i = threadIdx.x; i < iwCount; i += BLKT) {
        const float v = initW[i];
        s += v * v;
    }
    red[threadIdx.x] = s;
    __syncthreads();
    if (threadIdx.x == 0) {
        float nrm2 = 0.0f;
        for (int i = 0; i < BLKT; ++i) nrm2 += red[i];        // fixed order
        float data = 0.0f;
        for (int i = 0; i < nPart; ++i) data += partials[i];  // fixed order
        out[0] = data / (float)N_ROWS + REG_C * sqrtf(nrm2);
    }
}

// ---------------------------------------------------------------------------
extern "C" void kernel_launch(void* const* d_in, const int* in_sizes, int n_in,
                              void* d_out, int out_size, void* d_ws, size_t ws_size,
                              hipStream_t stream) {
    (void)n_in; (void)out_size; (void)ws_size;
    const float* W  = (const float*)d_in[0];   // weights      [C, D] f32
    const float* X  = (const float*)d_in[1];   // X            [N, D] f32
    const int*   Y  = (const int*)  d_in[2];   // labels       [N]    int
    const float* IW = (const float*)d_in[3];   // init_weights [C, D] f32
    float* out = (float*)d_out;

    float* Bp       = (float*)d_ws;            // packed B operands
    float* partials = Bp + PACK_FLOATS;        // per-block partials

    pack_weights_kernel<<<(PACK_FLOATS + BLKT - 1) / BLKT, BLKT, 0, stream>>>(W, Bp);
    logistic_loss_main_kernel<<<MAIN_BLOCKS, BLKT, 0, stream>>>(X, Bp, Y, partials);
    finalize_kernel<<<1, BLKT, 0, stream>>>(partials, MAIN_BLOCKS, IW,
                                            in_sizes[3], out);
}